// SpecConv_10290741641998
// MI455X (gfx1250) — compile-verified
//
#include <hip/hip_runtime.h>
#include <math.h>

typedef __attribute__((ext_vector_type(2))) float v2f;
typedef __attribute__((ext_vector_type(4))) float v4f;
typedef __attribute__((ext_vector_type(8))) float v8f;

#define N_TOT   100000
#define KDIM    128
#define INC     128
#define OUTC    128
#define STRIP   64
#define PBLK    128                       // # spec partial blocks
#define NSTRIPS ((N_TOT + STRIP - 1) / STRIP)   // 1563

__device__ __forceinline__ v8f v8f_zero() {
    v8f r;
#pragma unroll
    for (int i = 0; i < 8; ++i) r[i] = 0.0f;
    return r;
}

// ---- CDNA5 async copy helpers (ASYNCcnt-tracked, CDNA5 ISA) -------------
__device__ __forceinline__ void wait_async_0() {
    asm volatile("s_wait_asynccnt 0x0" ::: "memory");
}
__device__ __forceinline__ void wait_async_16() {   // allow next strip's 16 ops
    asm volatile("s_wait_asynccnt 0x10" ::: "memory");
}

// stage 64 rows x 128 f32 -> LDS (row stride 128), 8 x b128 per thread
__device__ __forceinline__ void stage_async128(const float* __restrict__ g,
                                               unsigned lds_byte_base,
                                               int base, int tid) {
#pragma unroll
    for (int j = 0; j < 8; ++j) {
        const int f   = tid + j * 256;
        const int row = f >> 5;
        const int c4  = (f & 31) << 2;
        const unsigned lds_addr = lds_byte_base + (unsigned)((row * 128 + c4) * 4);
        const unsigned long long ga =
            (unsigned long long)(uintptr_t)(g + (size_t)(base + row) * 128 + c4);
        asm volatile("global_load_async_to_lds_b128 %0, %1, off"
                     :: "v"(lds_addr), "v"(ga) : "memory");
    }
}

// same but padded LDS row stride 132 (conflict-free column reads)
__device__ __forceinline__ void stage_async_pad132(const float* __restrict__ g,
                                                   unsigned lds_byte_base,
                                                   int base, int tid) {
#pragma unroll
    for (int j = 0; j < 8; ++j) {
        const int f   = tid + j * 256;
        const int row = f >> 5;
        const int c4  = (f & 31) << 2;
        const unsigned lds_addr = lds_byte_base + (unsigned)((row * 132 + c4) * 4);
        const unsigned long long ga =
            (unsigned long long)(uintptr_t)(g + (size_t)(base + row) * 128 + c4);
        asm volatile("global_load_async_to_lds_b128 %0, %1, off"
                     :: "v"(lds_addr), "v"(ga) : "memory");
    }
}

// synchronous guarded staging for the tail strip (zero-fill OOB rows)
__device__ __forceinline__ void stage_sync128(const float* __restrict__ g,
                                              float* l, int base, int tid) {
#pragma unroll
    for (int j = 0; j < 8; ++j) {
        const int f   = tid + j * 256;
        const int row = f >> 5;
        const int c4  = (f & 31) << 2;
        const int n   = base + row;
        v4f u;
        if (n < N_TOT) u = *(const v4f*)(g + (size_t)n * 128 + c4);
        else { u[0] = u[1] = u[2] = u[3] = 0.0f; }
        *(v4f*)(l + row * 128 + c4) = u;
    }
}

// ---------------------------------------------------------------------------
// Kernel 1: spec partials.  spec[k,i] = sum_n U[n,k] * x[n,i]
// WMMA mapping: M=k (16), N=i (16), K=n (4).  Double-buffered async-to-LDS
// pipeline: DMA strip s+1 while WMMAs consume strip s.
// Buffer b: bytes [b*65536, +64KB) = U strip (32KB) then x strip (32KB).
// ---------------------------------------------------------------------------
__global__ __launch_bounds__(256) void spec_partial_kernel(
        const float* __restrict__ U, const float* __restrict__ x,
        float* __restrict__ partial) {
    __shared__ float lds[2 * 2 * STRIP * 128];          // 128 KB

    const int tid  = threadIdx.x;
    const int wave = tid >> 5;
    const int lane = tid & 31;
    const int l16  = lane & 15;
    const int kd0  = (lane >> 4) << 1;   // K-dim (n) sub-offset: 0 or 2
    const int k0   = wave * 16;          // this wave's k-tile
    const unsigned ldsBase = (unsigned)(uintptr_t)(void*)lds;

    v8f acc[8];
#pragma unroll
    for (int t = 0; t < 8; ++t) acc[t] = v8f_zero();

    int s   = blockIdx.x;
    int cur = 0;
    if (s < NSTRIPS) {                                   // prefetch first strip
        if (s * STRIP + STRIP <= N_TOT) {
            stage_async128(U, ldsBase,           s * STRIP, tid);
            stage_async128(x, ldsBase + 32768u,  s * STRIP, tid);
        } else {
            stage_sync128(U, lds,        s * STRIP, tid);
            stage_sync128(x, lds + 8192, s * STRIP, tid);
        }
    }

    while (s < NSTRIPS) {
        const int sn = s + (int)gridDim.x;
        int pend_next = 0;
        if (sn < NSTRIPS) {                              // issue next strip DMA
            const int nb = 1 - cur;
            if (sn * STRIP + STRIP <= N_TOT) {
                stage_async128(U, ldsBase + nb * 65536u,          sn * STRIP, tid);
                stage_async128(x, ldsBase + nb * 65536u + 32768u, sn * STRIP, tid);
                pend_next = 16;
            } else {
                stage_sync128(U, lds + nb * 16384,        sn * STRIP, tid);
                stage_sync128(x, lds + nb * 16384 + 8192, sn * STRIP, tid);
            }
        }
        // async completes in order: <=16 outstanding => current strip resident
        if (pend_next) wait_async_16(); else wait_async_0();
        __syncthreads();

        const float* lU = lds + cur * 16384;
        const float* lX = lU + 8192;
        for (int nc = 0; nc < STRIP; nc += 4) {
            const int r0 = nc + kd0;
            v2f A;
            A[0] = lU[r0 * 128 + k0 + l16];
            A[1] = lU[(r0 + 1) * 128 + k0 + l16];
#pragma unroll
            for (int t = 0; t < 8; ++t) {
                v2f B;
                B[0] = lX[r0 * 128 + t * 16 + l16];
                B[1] = lX[(r0 + 1) * 128 + t * 16 + l16];
                acc[t] = __builtin_amdgcn_wmma_f32_16x16x4_f32(
                    false, A, false, B, (short)0, acc[t], false, false);
            }
        }
        __syncthreads();                                 // buf reusable
        cur ^= 1;
        s = sn;
    }

    // write this block's 128x128 partial (D layout: vgpr j -> row j / j+8)
    float* P = partial + (size_t)blockIdx.x * (KDIM * INC);
    const int roff = (lane >> 4) * 8;
#pragma unroll
    for (int t = 0; t < 8; ++t)
#pragma unroll
        for (int j = 0; j < 8; ++j)
            P[(k0 + j + roff) * INC + t * 16 + l16] = acc[t][j];
}

// ---------------------------------------------------------------------------
// Kernel 2: fixed-order reduction of partials -> spec (deterministic)
// ---------------------------------------------------------------------------
__global__ void spec_reduce_kernel(const float* __restrict__ partial,
                                   float* __restrict__ spec) {
    const int idx = blockIdx.x * blockDim.x + threadIdx.x;  // 16384
    float s = 0.0f;
    for (int p = 0; p < PBLK; ++p)
        s += partial[(size_t)p * (KDIM * INC) + idx];
    spec[idx] = s;
}

// ---------------------------------------------------------------------------
// Kernel 3: mixed[o,k] = dot(coeffs[o,k,:], spec[k,:]); packed layout
// mixedP[(k>>1)*256 + o*2 + (k&1)] so kernel-4 B fragments are float2 loads.
// ---------------------------------------------------------------------------
__global__ void mix_kernel(const float* __restrict__ coeffs,
                           const float* __restrict__ spec,
                           float* __restrict__ mixedP) {
    const int idx = blockIdx.x * blockDim.x + threadIdx.x;  // 16384
    const int o = idx >> 7, k = idx & 127;
    const v4f* c  = (const v4f*)(coeffs + (size_t)idx * 128);
    const v4f* sp = (const v4f*)(spec + k * 128);
    float s = 0.0f;
#pragma unroll 4
    for (int i = 0; i < 32; ++i) {
        v4f a = c[i], b = sp[i];
        s += a[0] * b[0] + a[1] * b[1] + a[2] * b[2] + a[3] * b[3];
    }
    mixedP[(k >> 1) * 256 + o * 2 + (k & 1)] = s;
}

// ---------------------------------------------------------------------------
// Kernel 4: z[n,o] = sum_k U[n,k] * mixed[o,k]  (M=n, N=o, K=k)
// Async-staged U strip in LDS (stride 132 -> conflict-free column reads).
// 8 waves = 4 n-tiles x 2 o-halves.  z lands in d_out, softmax runs in place.
// ---------------------------------------------------------------------------
__global__ __launch_bounds__(256) void out_gemm_kernel(
        const float* __restrict__ U, const float* __restrict__ mixedP,
        float* __restrict__ z) {
    __shared__ float ldsU[STRIP * 132];

    const int tid   = threadIdx.x;
    const int wave  = tid >> 5;
    const int lane  = tid & 31;
    const int l16   = lane & 15;
    const int kd0   = (lane >> 4) << 1;
    const int ntile = wave & 3;
    const int ohalf = wave >> 2;
    const int base  = blockIdx.x * STRIP;

    if (base + STRIP <= N_TOT) {
        stage_async_pad132(U, (unsigned)(uintptr_t)(void*)ldsU, base, tid);
        wait_async_0();
    } else {
#pragma unroll
        for (int j = 0; j < 8; ++j) {
            const int f   = tid + j * 256;
            const int row = f >> 5;
            const int c4  = (f & 31) << 2;
            const int n   = base + row;
            v4f u;
            if (n < N_TOT) u = *(const v4f*)(U + (size_t)n * 128 + c4);
            else { u[0] = u[1] = u[2] = u[3] = 0.0f; }
            *(v4f*)(ldsU + row * 132 + c4) = u;
        }
    }
    __syncthreads();

    v8f acc[4];
#pragma unroll
    for (int t = 0; t < 4; ++t) acc[t] = v8f_zero();

    for (int kc = 0; kc < KDIM; kc += 4) {
        const int kk = kc + kd0;                         // even
        const v2f A = *(const v2f*)(ldsU + (ntile * 16 + l16) * 132 + kk);
#pragma unroll
        for (int t = 0; t < 4; ++t) {
            const int o0 = (ohalf * 4 + t) * 16;
            const v2f B = *(const v2f*)(mixedP + (kk >> 1) * 256 + (o0 + l16) * 2);
            acc[t] = __builtin_amdgcn_wmma_f32_16x16x4_f32(
                false, A, false, B, (short)0, acc[t], false, false);
        }
    }

    const int roff = (lane >> 4) * 8;
#pragma unroll
    for (int t = 0; t < 4; ++t) {
        const int o0 = (ohalf * 4 + t) * 16;
#pragma unroll
        for (int j = 0; j < 8; ++j) {
            const int n = base + ntile * 16 + j + roff;
            if (n < N_TOT) z[(size_t)n * 128 + o0 + l16] = acc[t][j];
        }
    }
}

// ---------------------------------------------------------------------------
// Kernel 5: per-channel softmax stats (max, 1/sum(exp)); fixed tree order.
// ---------------------------------------------------------------------------
__global__ __launch_bounds__(256) void softmax_stats_kernel(
        const float* __restrict__ z, float* __restrict__ stats) {
    __shared__ float red[256];
    const int o = blockIdx.x, tid = threadIdx.x;

    float m = -3.402823466e38f;
    for (int n = tid; n < N_TOT; n += 256)
        m = fmaxf(m, z[(size_t)n * 128 + o]);
    red[tid] = m;
    __syncthreads();
    for (int s = 128; s > 0; s >>= 1) {
        if (tid < s) red[tid] = fmaxf(red[tid], red[tid + s]);
        __syncthreads();
    }
    const float mx = red[0];
    __syncthreads();

    float sum = 0.0f;
    for (int n = tid; n < N_TOT; n += 256)
        sum += expf(z[(size_t)n * 128 + o] - mx);
    red[tid] = sum;
    __syncthreads();
    for (int s = 128; s > 0; s >>= 1) {
        if (tid < s) red[tid] += red[tid + s];
        __syncthreads();
    }
    if (tid == 0) { stats[o] = mx; stats[128 + o] = 1.0f / red[0]; }
}

// ---------------------------------------------------------------------------
// Kernel 6: in-place normalization of d_out
// ---------------------------------------------------------------------------
__global__ void normalize_kernel(float* __restrict__ z,
                                 const float* __restrict__ stats) {
    const size_t idx = (size_t)blockIdx.x * 256 + threadIdx.x;
    if (idx < (size_t)N_TOT * 128) {
        const int o = (int)(idx & 127);
        z[idx] = expf(z[idx] - stats[o]) * stats[128 + o];
    }
}

// ---------------------------------------------------------------------------
extern "C" void kernel_launch(void* const* d_in, const int* in_sizes, int n_in,
                              void* d_out, int out_size, void* d_ws, size_t ws_size,
                              hipStream_t stream) {
    const float* x      = (const float*)d_in[0];   // (N, 128)
    const float* U      = (const float*)d_in[1];   // (N, 128)
    const float* coeffs = (const float*)d_in[2];   // (128, 128, 128)
    float* out = (float*)d_out;                    // (N, 128)

    float* ws      = (float*)d_ws;
    float* partial = ws;                                   // PBLK*16384 f32 (8 MB)
    float* spec    = ws + (size_t)PBLK * (KDIM * INC);     // 16384 f32
    float* mixedP  = spec + KDIM * INC;                    // 16384 f32
    float* stats   = mixedP + OUTC * KDIM;                 // 256 f32

    spec_partial_kernel<<<PBLK, 256, 0, stream>>>(U, x, partial);
    spec_reduce_kernel<<<(KDIM * INC) / 256, 256, 0, stream>>>(partial, spec);
    mix_kernel<<<(OUTC * KDIM) / 256, 256, 0, stream>>>(coeffs, spec, mixedP);
    out_gemm_kernel<<<NSTRIPS, 256, 0, stream>>>(U, mixedP, out);
    softmax_stats_kernel<<<OUTC, 256, 0, stream>>>(out, stats);
    normalize_kernel<<<(int)(((size_t)N_TOT * 128 + 255) / 256), 256, 0, stream>>>(out, stats);
}